// RNNLayer_8959301780016
// MI455X (gfx1250) — compile-verified
//
#include <hip/hip_runtime.h>
#include <math.h>

// Problem constants (from the reference)
#define SEQ   784
#define BS    256
#define DH    512
#define PAD   16    // trailing zeros after the reversed tap array

typedef __attribute__((ext_vector_type(2))) float v2f;
typedef __attribute__((ext_vector_type(8))) float v8f;

// ---------------------------------------------------------------------------
// Kernel 1: reversed taps.
//   kr[i]          = k[SEQ-1-i] = sum_j b[j]*c[j]*a[j]^(SEQ-1-i),  i < SEQ
//   kr[SEQ..SEQ+PAD-1] = 0      (covers the d<0 region of the diagonal tiles)
// One thread per entry, no barriers. a ~ 1.0 so a^d = exp2(d*log2 a) is
// ~1e-6-accurate in fp32.
// ---------------------------------------------------------------------------
__global__ void __launch_bounds__(256)
rnn_taps_kernel(const float* __restrict__ a, const float* __restrict__ b,
                const float* __restrict__ c, float* __restrict__ kr) {
    const int idx = blockIdx.x * 256 + threadIdx.x;
    if (idx >= SEQ + PAD) return;
    if (idx >= SEQ) { kr[idx] = 0.0f; return; }
    const float d = (float)(SEQ - 1 - idx);
    float acc = 0.0f;
    for (int j = 0; j < DH; ++j) {
        acc += b[j] * c[j] * exp2f(d * log2f(a[j]));
    }
    kr[idx] = acc;
}

// ---------------------------------------------------------------------------
// Kernel 2: Y[t][n] = sum_{s<=t} k[t-s] * X[s][n] as a Toeplitz GEMM on
// V_WMMA_F32_16X16X4_F32 (full fp32 precision).
//
// One 16x16 output tile per wave; 49 x 16 = 784 tiles, 8 waves/block,
// 98 blocks (exact). Loop trip count is made explicitly wave-uniform
// (scalar) via readfirstlane so EXEC is never touched around the WMMAs.
//
// ISA layouts (CDNA5 ISA):
//   A 16x4 f32 : lanes 0-15 -> M=lane (K base+0/+1 in v0/v1),
//                lanes 16-31 -> same M, K base+2/+3.
//   B 4x16 f32 : symmetric (N across lanes in half-wave, K split likewise).
//   C/D 16x16  : VGPR r -> M=r (lanes 0-15), M=r+8 (lanes 16-31); N = lane&15.
//
// A.x = k[t-s] = kr[SEQ-1-(t-s)], A.y = k[t-s-1] = kr[SEQ-(t-s)] are
// adjacent ascending addresses -> single b64 load, no register swap.
// ---------------------------------------------------------------------------
__global__ void __launch_bounds__(256)
rnn_toeplitz_wmma_kernel(const float* __restrict__ kr,
                         const float* __restrict__ x,
                         float* __restrict__ y) {
    const int wave = __builtin_amdgcn_readfirstlane((int)(threadIdx.x >> 5));
    const int lane = threadIdx.x & 31;
    const int tile = blockIdx.x * 8 + wave;   // fully scalar

    const int tr = tile >> 4;   // tile row  0..48  (scalar)
    const int tc = tile & 15;   // tile col  0..15  (scalar)

    const int t0   = tr * 16;
    const int n0   = tc * 16;
    const int half = lane >> 4; // 0: lanes 0-15, 1: lanes 16-31
    const int l    = lane & 15;

    // Per-lane base pointers at s0 = 0, kk = 0 (kA0 = half*2):
    //   A index: SEQ-1 - (t0+l) + kA0   (grows by +4 per kk, +16 per chunk)
    //   B addr : x[kA0*BS + n0 + l]
    const float* pA = kr + (SEQ - 1 - t0 - l) + half * 2;
    const float* pB = x + (size_t)(half * 2) * BS + n0 + l;

    v8f acc = {};

    for (int cc = 0; cc <= tr; ++cc) {        // scalar loop control
        #pragma unroll
        for (int kk = 0; kk < 16; kk += 4) {
            v2f A, B;
            A.x = pA[kk];                     // k[t-s]   (or trailing zero)
            A.y = pA[kk + 1];                 // k[t-s-1]
            B.x = pB[(size_t)kk * BS];        // X[s][n]
            B.y = pB[(size_t)(kk + 1) * BS];  // X[s+1][n]
            acc = __builtin_amdgcn_wmma_f32_16x16x4_f32(
                /*neg_a=*/false, A, /*neg_b=*/false, B,
                /*c_mod=*/(short)0, acc, /*reuse_a=*/false, /*reuse_b=*/false);
        }
        pA += 16;
        pB += (size_t)16 * BS;
    }

    // Store D per 16x16 f32 C/D layout.
    float* py = y + (size_t)(t0 + half * 8) * BS + n0 + l;
    #pragma unroll
    for (int r = 0; r < 8; ++r) {
        py[(size_t)r * BS] = acc[r];
    }
}

// ---------------------------------------------------------------------------
// Launch: inputs in setup_inputs() order: x, a, b, c (all fp32).
// d_out: SEQ*BS fp32. d_ws: SEQ+PAD floats of scratch.
// ---------------------------------------------------------------------------
extern "C" void kernel_launch(void* const* d_in, const int* in_sizes, int n_in,
                              void* d_out, int out_size, void* d_ws, size_t ws_size,
                              hipStream_t stream) {
    const float* x = (const float*)d_in[0];
    const float* a = (const float*)d_in[1];
    const float* b = (const float*)d_in[2];
    const float* c = (const float*)d_in[3];
    float* y  = (float*)d_out;
    float* kr = (float*)d_ws;   // SEQ + PAD floats of scratch

    const int ntaps = SEQ + PAD;                    // 800
    rnn_taps_kernel<<<(ntaps + 255) / 256, 256, 0, stream>>>(a, b, c, kr);

    const int tiles  = (SEQ / 16) * (BS / 16); // 784
    const int blocks = tiles / 8;              // 98
    rnn_toeplitz_wmma_kernel<<<blocks, 256, 0, stream>>>(kr, x, y);
}